// structural_Des_80985903334292
// MI455X (gfx1250) — compile-verified
//
#include <hip/hip_runtime.h>
#include <hip/hip_bf16.h>
#include <math.h>

// ---------------------------------------------------------------------------
// MI455X (gfx1250) implementation.
// Roofline: ~22 GFLOP vs ~425 MB mandatory HBM traffic -> memory bound
// (~18us @ 23.3 TB/s). All four GEMMs of the pipeline that matter (lin3,
// lin4, conv1, conv2) run on V_WMMA_F32_16X16X4_F32 (fp32 for accuracy);
// the vector pipe only does the cheap per-point prolog + exp-RBF. Bulk HBM
// traffic (h stage + output) moves as coalesced b128 via LDS tiles.
// ---------------------------------------------------------------------------

typedef __attribute__((ext_vector_type(2))) float v2f;
typedef __attribute__((ext_vector_type(4))) float v4f;
typedef __attribute__((ext_vector_type(8))) float v8f;

#define NPTS 262144
#define ROWS 64             // points per block
#define NBLK (NPTS / ROWS)  // 4096
#define CH   131
#define KPAD 132            // K padded to multiple of 4 (GEMM1/2)
#define NPAD 144            // N padded to multiple of 16 (GEMM1/2)
#define LDW  132            // LDS row stride for 131-wide tiles
#define ADW  34             // LDS row stride for the 32-wide v0 tile
#define TDW  66             // LDS row stride for the 64-wide t tile

#define HTILE (ROWS * LDW)  // 8448 floats per block, contiguous in hbuf

// workspace layout (in floats)
#define WP1_OFF    0u
#define WP2_OFF    (WP1_OFF + (unsigned)(KPAD * NPAD))   // 19008
#define WP3_OFF    (WP2_OFF + (unsigned)(KPAD * NPAD))   // 38016
#define WP4_OFF    (WP3_OFF + 2048u)                     // 40064
#define WC_OFF     (WP4_OFF + 4096u)                     // 44160
#define KL2E_OFF   (WC_OFF + 128u)                       // 44288
#define C2_OFF     (KL2E_OFF + 768u)                     // 45056
#define SCALE_OFF  (C2_OFF + 256u)                       // 45312
#define SHIFT_OFF  (SCALE_OFF + 256u)
#define PART_OFF   (SHIFT_OFF + 256u)
#define PARTSQ_OFF (PART_OFF + (unsigned)(NBLK * CH))
#define H_OFF      (PARTSQ_OFF + (unsigned)(NBLK * CH))

__device__ __forceinline__ v8f wmma_f32_16x16x4(v2f a, v2f b, v8f c) {
  // 8 args: (neg_a, A, neg_b, B, c_mod, C, reuse_a, reuse_b)
  return __builtin_amdgcn_wmma_f32_16x16x4_f32(false, a, false, b, (short)0, c,
                                               false, false);
}

// Build padded/transposed B-matrices for GEMM1/2: Wp[k][n] = W[n][k].
__global__ __launch_bounds__(256) void prep_weights(const float* __restrict__ w1,
                                                    const float* __restrict__ w2,
                                                    float* __restrict__ wp1,
                                                    float* __restrict__ wp2) {
  int idx = blockIdx.x * blockDim.x + threadIdx.x;
  const int tot = KPAD * NPAD;
  if (idx >= 2 * tot) return;
  const float* w = (idx < tot) ? w1 : w2;
  float* wp = (idx < tot) ? wp1 : wp2;
  int r = (idx < tot) ? idx : idx - tot;
  int k = r / NPAD, n = r % NPAD;
  wp[r] = (k < CH && n < CH) ? w[n * CH + k] : 0.0f;
}

// Small precomputation: lin3/lin4 B-matrices, collapsed conv weights, and
// RBF tables in exp2 form.
__global__ __launch_bounds__(256) void prep_small(
    const float* __restrict__ conv_w, const float* __restrict__ lin3_w,
    const float* __restrict__ lin4_w, const float* __restrict__ kern,
    float* __restrict__ wp3, float* __restrict__ wp4, float* __restrict__ wc,
    float* __restrict__ kl2e, float* __restrict__ c2) {
  const int t = threadIdx.x;
  const float L2E = 1.4426950408889634f;  // log2(e)
  for (int i = t; i < 32 * 64; i += 256) {  // wp3[k][n] = lin3_w[n][k]
    int k = i / 64, n = i % 64;
    wp3[i] = lin3_w[n * 32 + k];
  }
  for (int i = t; i < 64 * 64; i += 256) {  // wp4[k][n] = lin4_w[n][k]
    int k = i / 64, n = i % 64;
    wp4[i] = lin4_w[n * 64 + k];
  }
  for (int i = t; i < 96; i += 256) {  // wc[o][d] = (wA+wB)/3
    int o = i / 3, d = i % 3;
    wc[i] = (conv_w[o * 6 + d] + conv_w[o * 6 + 3 + d]) * (1.0f / 3.0f);
  }
  for (int i = t; i < 256; i += 256) {  // (m,l) flattened
    float kx = kern[i * 3 + 0], ky = kern[i * 3 + 1], kz = kern[i * 3 + 2];
    kl2e[i * 3 + 0] = 2.0f * L2E * kx;
    kl2e[i * 3 + 1] = 2.0f * L2E * ky;
    kl2e[i * 3 + 2] = 2.0f * L2E * kz;
    c2[i] = -L2E * (kx * kx + ky * ky + kz * kz);
  }
}

// Fused: per-point prolog (VALU) -> lin3/lin4 GEMMs (WMMA) -> x row in LDS
// -> GEMM1 (WMMA) -> h tile to workspace (b128) + per-block BN partials.
__global__ __launch_bounds__(64) void fused_feat_gemm1(
    const float* __restrict__ corner, const float* __restrict__ normal,
    const float* __restrict__ wc, const float* __restrict__ conv_b,
    const float* __restrict__ wp3, const float* __restrict__ lin3_b,
    const float* __restrict__ wp4, const float* __restrict__ lin4_b,
    const float* __restrict__ kl2e, const float* __restrict__ c2,
    const float* __restrict__ wp1, const float* __restrict__ conv1_b,
    float* __restrict__ hbuf, float* __restrict__ part,
    float* __restrict__ partsq) {
  __shared__ __align__(16) float xt[HTILE];        // 33,792 B
  __shared__ __align__(16) float at[ROWS * ADW];   //  8,704 B (v0 tile)
  __shared__ __align__(16) float tt[ROWS * TDW];   // 16,896 B (t tile)

  const int tid = threadIdx.x;
  const int blk = blockIdx.x;
  const int n = blk * ROWS + tid;
  const float L2E = 1.4426950408889634f;

  // ---------------- phase 1: per-point prolog (cheap VALU) -----------------
  {
    float cr[9];
#pragma unroll
    for (int i = 0; i < 9; ++i) cr[i] = corner[(size_t)n * 9 + i];
    // mean over the 3 pairs collapses to wc . (c1+c2+c3)
    float s0 = cr[0] + cr[3] + cr[6];
    float s1 = cr[1] + cr[4] + cr[7];
    float s2 = cr[2] + cr[5] + cr[8];
    float nx = normal[(size_t)n * 3 + 0];
    float ny = normal[(size_t)n * 3 + 1];
    float nz = normal[(size_t)n * 3 + 2];
    float nn = nx * nx + ny * ny + nz * nz;

    float* arow = at + tid * ADW;
#pragma unroll
    for (int o = 0; o < 32; ++o)
      arow[o] = conv_b[o] + fmaf(wc[o * 3 + 0], s0,
                  fmaf(wc[o * 3 + 1], s1, wc[o * 3 + 2] * s2));

    float* row = xt + tid * LDW;
    row[0] = nx; row[1] = ny; row[2] = nz;
    // RBF z -> cols 67..130:
    //   exp(-(nn + |k|^2 - 2 n.k)) = 2^(-L2E*nn) * 2^(L2E*(2 n.k) - L2E*|k|^2)
    float en = exp2f(-L2E * nn) * (1.0f / 32.0f);  // includes 0.5/(K*4)
    for (int m = 0; m < 64; ++m) {
      float zz = 0.0f;
#pragma unroll
      for (int l = 0; l < 4; ++l) {
        int idx = m * 4 + l;
        float e = fmaf(kl2e[idx * 3 + 0], nx,
                   fmaf(kl2e[idx * 3 + 1], ny,
                    fmaf(kl2e[idx * 3 + 2], nz, c2[idx])));
        zz += exp2f(e);
      }
      row[67 + m] = zz * en;
    }
    row[131] = 0.0f;  // K pad column (stays 0 through phase 4)
  }
  __syncthreads();

  const int lane = tid & 31;
  const int waveM = (tid >> 5) * 2;  // wave0 -> M tiles 0,1 ; wave1 -> 2,3
  const int klane = (lane >> 4) * 2;
  const int nlane = lane & 15;
  const int rofs = (lane >= 16) ? 8 : 0;

  // ---------------- phase 2a: t = relu(v0 @ lin3_w^T + b)  (WMMA) ----------
  {
    v8f acc[2][4];
#pragma unroll
    for (int mt = 0; mt < 2; ++mt)
#pragma unroll
      for (int nt = 0; nt < 4; ++nt)
#pragma unroll
        for (int i = 0; i < 8; ++i) acc[mt][nt][i] = 0.0f;

    for (int k = 0; k < 32; k += 4) {
      v2f a[2];
#pragma unroll
      for (int mt = 0; mt < 2; ++mt) {
        int ar = (waveM + mt) * 16 + nlane;
        a[mt][0] = at[ar * ADW + k + klane + 0];
        a[mt][1] = at[ar * ADW + k + klane + 1];
      }
#pragma unroll
      for (int nt = 0; nt < 4; ++nt) {
        v2f b;
        b[0] = wp3[(k + klane + 0) * 64 + nt * 16 + nlane];
        b[1] = wp3[(k + klane + 1) * 64 + nt * 16 + nlane];
#pragma unroll
        for (int mt = 0; mt < 2; ++mt)
          acc[mt][nt] = wmma_f32_16x16x4(a[mt], b, acc[mt][nt]);
      }
    }
    __syncthreads();
#pragma unroll
    for (int mt = 0; mt < 2; ++mt)
#pragma unroll
      for (int nt = 0; nt < 4; ++nt)
#pragma unroll
        for (int i = 0; i < 8; ++i) {
          int r = (waveM + mt) * 16 + i + rofs;
          int c = nt * 16 + nlane;
          tt[r * TDW + c] = fmaxf(acc[mt][nt][i] + lin3_b[c], 0.0f);
        }
  }
  __syncthreads();

  // ---------------- phase 2b: v = t @ lin4_w^T + b  (WMMA) -----------------
  {
    v8f acc[2][4];
#pragma unroll
    for (int mt = 0; mt < 2; ++mt)
#pragma unroll
      for (int nt = 0; nt < 4; ++nt)
#pragma unroll
        for (int i = 0; i < 8; ++i) acc[mt][nt][i] = 0.0f;

    for (int k = 0; k < 64; k += 4) {
      v2f a[2];
#pragma unroll
      for (int mt = 0; mt < 2; ++mt) {
        int ar = (waveM + mt) * 16 + nlane;
        a[mt][0] = tt[ar * TDW + k + klane + 0];
        a[mt][1] = tt[ar * TDW + k + klane + 1];
      }
#pragma unroll
      for (int nt = 0; nt < 4; ++nt) {
        v2f b;
        b[0] = wp4[(k + klane + 0) * 64 + nt * 16 + nlane];
        b[1] = wp4[(k + klane + 1) * 64 + nt * 16 + nlane];
#pragma unroll
        for (int mt = 0; mt < 2; ++mt)
          acc[mt][nt] = wmma_f32_16x16x4(a[mt], b, acc[mt][nt]);
      }
    }
#pragma unroll
    for (int mt = 0; mt < 2; ++mt)
#pragma unroll
      for (int nt = 0; nt < 4; ++nt)
#pragma unroll
        for (int i = 0; i < 8; ++i) {
          int r = (waveM + mt) * 16 + i + rofs;
          int c = nt * 16 + nlane;
          xt[r * LDW + 3 + c] = acc[mt][nt][i] + lin4_b[c];  // v -> cols 3..66
        }
  }
  __syncthreads();

  // ---------------- phase 3: h = x @ conv1_w^T  (WMMA) ---------------------
  v8f acc[2][9];
#pragma unroll
  for (int mt = 0; mt < 2; ++mt)
#pragma unroll
    for (int nt = 0; nt < 9; ++nt)
#pragma unroll
      for (int i = 0; i < 8; ++i) acc[mt][nt][i] = 0.0f;

  for (int k = 0; k < KPAD; k += 4) {
    v2f a[2];
#pragma unroll
    for (int mt = 0; mt < 2; ++mt) {
      int ar = (waveM + mt) * 16 + nlane;
      a[mt][0] = xt[ar * LDW + k + klane + 0];
      a[mt][1] = xt[ar * LDW + k + klane + 1];
    }
#pragma unroll
    for (int nt = 0; nt < 9; ++nt) {
      v2f b;
      b[0] = wp1[(k + klane + 0) * NPAD + nt * 16 + nlane];
      b[1] = wp1[(k + klane + 1) * NPAD + nt * 16 + nlane];
#pragma unroll
      for (int mt = 0; mt < 2; ++mt)
        acc[mt][nt] = wmma_f32_16x16x4(a[mt], b, acc[mt][nt]);
    }
  }
  __syncthreads();

  // ---------------- phase 4: D tiles (+bias) -> LDS ------------------------
#pragma unroll
  for (int mt = 0; mt < 2; ++mt)
#pragma unroll
    for (int nt = 0; nt < 9; ++nt)
#pragma unroll
      for (int i = 0; i < 8; ++i) {
        int r = (waveM + mt) * 16 + i + rofs;
        int c = nt * 16 + nlane;
        if (c < CH) xt[r * LDW + c] = acc[mt][nt][i] + conv1_b[c];
      }
  __syncthreads();

  // ---------------- phase 5: h tile -> hbuf (b128) + partials --------------
  {
    const v4f* xs = (const v4f*)xt;
    v4f* hb = (v4f*)(hbuf + (size_t)blk * HTILE);
    for (int i = tid; i < HTILE / 4; i += ROWS) hb[i] = xs[i];
  }
  for (int c = tid; c < CH; c += ROWS) {
    float s = 0.0f, sq = 0.0f;
    for (int r = 0; r < ROWS; ++r) {
      float v = xt[r * LDW + c];
      s += v;
      sq = fmaf(v, v, sq);
    }
    part[(size_t)blk * CH + c] = s;
    partsq[(size_t)blk * CH + c] = sq;
  }
}

// Per-channel reduction of block partials -> BN scale/shift.
__global__ __launch_bounds__(256) void bn_reduce(const float* __restrict__ part,
                                                 const float* __restrict__ partsq,
                                                 const float* __restrict__ bn_g,
                                                 const float* __restrict__ bn_b,
                                                 float* __restrict__ scale,
                                                 float* __restrict__ shift) {
  __shared__ float r1[256];
  __shared__ float r2[256];
  const int ch = blockIdx.x;
  const int t = threadIdx.x;
  float a = 0.0f, b = 0.0f;
  for (int i = t; i < NBLK; i += 256) {
    a += part[(size_t)i * CH + ch];
    b += partsq[(size_t)i * CH + ch];
  }
  r1[t] = a; r2[t] = b;
  __syncthreads();
  for (int off = 128; off > 0; off >>= 1) {
    if (t < off) { r1[t] += r1[t + off]; r2[t] += r2[t + off]; }
    __syncthreads();
  }
  if (t == 0) {
    float mean = r1[0] * (1.0f / (float)NPTS);
    float var = r2[0] * (1.0f / (float)NPTS) - mean * mean;
    float inv = rsqrtf(var + 1e-5f);
    float sc = bn_g[ch] * inv;
    scale[ch] = sc;
    shift[ch] = bn_b[ch] - mean * sc;
  }
}

// out = relu(BN(h)) @ conv2_w^T + conv2_b via WMMA f32; all global traffic
// staged through LDS as coalesced b128.
__global__ __launch_bounds__(64) void bn_gemm2(
    const float* __restrict__ hbuf, const float* __restrict__ scale,
    const float* __restrict__ shift, const float* __restrict__ wp2,
    const float* __restrict__ conv2_b, float* __restrict__ out) {
  __shared__ __align__(16) float xt[HTILE];
  const int tid = threadIdx.x;
  const int blk = blockIdx.x;

  // vectorized load of h tile with fused BN + relu (pad col -> 0)
  {
    const v4f* hb = (const v4f*)(hbuf + (size_t)blk * HTILE);
    v4f* xs = (v4f*)xt;
    for (int i = tid; i < HTILE / 4; i += ROWS) {
      v4f h = hb[i];
      int c0 = (i % (LDW / 4)) * 4;  // column of first element, multiple of 4
      v4f r;
#pragma unroll
      for (int j = 0; j < 4; ++j) {
        int c = c0 + j;
        float v = 0.0f;
        if (c < CH) v = fmaxf(fmaf(h[j], scale[c], shift[c]), 0.0f);
        r[j] = v;
      }
      xs[i] = r;
    }
  }
  __syncthreads();

  const int lane = tid & 31;
  const int waveM = (tid >> 5) * 2;
  const int klane = (lane >> 4) * 2;
  const int nlane = lane & 15;

  v8f acc[2][9];
#pragma unroll
  for (int mt = 0; mt < 2; ++mt)
#pragma unroll
    for (int nt = 0; nt < 9; ++nt)
#pragma unroll
      for (int i = 0; i < 8; ++i) acc[mt][nt][i] = 0.0f;

  for (int k = 0; k < KPAD; k += 4) {
    v2f a[2];
#pragma unroll
    for (int mt = 0; mt < 2; ++mt) {
      int ar = (waveM + mt) * 16 + nlane;
      a[mt][0] = xt[ar * LDW + k + klane + 0];
      a[mt][1] = xt[ar * LDW + k + klane + 1];
    }
#pragma unroll
    for (int nt = 0; nt < 9; ++nt) {
      v2f b;
      b[0] = wp2[(k + klane + 0) * NPAD + nt * 16 + nlane];
      b[1] = wp2[(k + klane + 1) * NPAD + nt * 16 + nlane];
#pragma unroll
      for (int mt = 0; mt < 2; ++mt)
        acc[mt][nt] = wmma_f32_16x16x4(a[mt], b, acc[mt][nt]);
    }
  }
  __syncthreads();

  // D tiles (+bias) -> LDS
  const int rofs = (lane >= 16) ? 8 : 0;
#pragma unroll
  for (int mt = 0; mt < 2; ++mt)
#pragma unroll
    for (int nt = 0; nt < 9; ++nt)
#pragma unroll
      for (int i = 0; i < 8; ++i) {
        int r = (waveM + mt) * 16 + i + rofs;
        int c = nt * 16 + nlane;
        if (c < CH) xt[r * LDW + c] = acc[mt][nt][i] + conv2_b[c];
      }
  __syncthreads();

  // block's 64 output rows are contiguous in global: coalesced b128 stores
  {
    v4f* o4 = (v4f*)(out + (size_t)blk * (ROWS * CH));
    for (int i = tid; i < (ROWS * CH) / 4; i += ROWS) {
      int base = i * 4;
      v4f r;
#pragma unroll
      for (int j = 0; j < 4; ++j) {
        int idx = base + j;
        int rr = idx / CH, cc = idx - rr * CH;
        r[j] = xt[rr * LDW + cc];
      }
      o4[i] = r;
    }
  }
}

extern "C" void kernel_launch(void* const* d_in, const int* in_sizes, int n_in,
                              void* d_out, int out_size, void* d_ws,
                              size_t ws_size, hipStream_t stream) {
  (void)in_sizes; (void)n_in; (void)out_size; (void)ws_size;
  const float* corner  = (const float*)d_in[0];
  const float* normal  = (const float*)d_in[1];
  // d_in[2] = neighbour (int64) is unused by the reference
  const float* conv_w  = (const float*)d_in[3];
  const float* conv_b  = (const float*)d_in[4];
  const float* lin3_w  = (const float*)d_in[5];
  const float* lin3_b  = (const float*)d_in[6];
  const float* lin4_w  = (const float*)d_in[7];
  const float* lin4_b  = (const float*)d_in[8];
  const float* kern    = (const float*)d_in[9];
  const float* conv1_w = (const float*)d_in[10];
  const float* conv1_b = (const float*)d_in[11];
  const float* bn_g    = (const float*)d_in[12];
  const float* bn_b    = (const float*)d_in[13];
  const float* conv2_w = (const float*)d_in[14];
  const float* conv2_b = (const float*)d_in[15];

  float* ws = (float*)d_ws;
  float* wp1   = ws + WP1_OFF;
  float* wp2   = ws + WP2_OFF;
  float* wp3   = ws + WP3_OFF;
  float* wp4   = ws + WP4_OFF;
  float* wc    = ws + WC_OFF;
  float* kl2e  = ws + KL2E_OFF;
  float* c2    = ws + C2_OFF;
  float* scale = ws + SCALE_OFF;
  float* shift = ws + SHIFT_OFF;
  float* part  = ws + PART_OFF;
  float* psq   = ws + PARTSQ_OFF;
  float* hbuf  = ws + H_OFF;

  prep_weights<<<(2 * KPAD * NPAD + 255) / 256, 256, 0, stream>>>(
      conv1_w, conv2_w, wp1, wp2);
  prep_small<<<1, 256, 0, stream>>>(conv_w, lin3_w, lin4_w, kern, wp3, wp4,
                                    wc, kl2e, c2);
  fused_feat_gemm1<<<NBLK, ROWS, 0, stream>>>(
      corner, normal, wc, conv_b, wp3, lin3_b, wp4, lin4_b, kl2e, c2,
      wp1, conv1_b, hbuf, part, psq);
  bn_reduce<<<CH, 256, 0, stream>>>(part, psq, bn_g, bn_b, scale, shift);
  bn_gemm2<<<NBLK, ROWS, 0, stream>>>(hbuf, scale, shift, wp2, conv2_b,
                                      (float*)d_out);
}